// GeometricEncoder_89781996355965
// MI455X (gfx1250) — compile-verified
//
#include <hip/hip_runtime.h>

#define N_NODES 100000
#define N_EDGES 1000000
#define HIDDEN  64
#define N_LAYERS 3

typedef __attribute__((ext_vector_type(2))) float v2f;
typedef __attribute__((ext_vector_type(8))) float v8f;

// Guaranteed-native CDNA5 global f32 atomic add (no CAS-loop fallback).
__device__ __forceinline__ void atomic_add_f32_global(float* p, float v) {
  asm volatile("global_atomic_add_f32 %0, %1, off" :: "v"(p), "v"(v) : "memory");
}

// ---------------- x = emb[z] ----------------
__global__ void embed_kernel(const int* __restrict__ z, const float* __restrict__ emb,
                             float* __restrict__ x) {
  int idx = blockIdx.x * blockDim.x + threadIdx.x;       // over N*64
  if (idx >= N_NODES * HIDDEN) return;
  int n = idx >> 6, h = idx & 63;
  x[idx] = emb[z[n] * HIDDEN + h];
}

// ---------------- rbf[e] = exp(-||pos[row]-pos[col]||) ----------------
__global__ void rbf_kernel(const float* __restrict__ pos, const int* __restrict__ row,
                           const int* __restrict__ col, float* __restrict__ rbf) {
  int e = blockIdx.x * blockDim.x + threadIdx.x;
  if (e >= N_EDGES) return;
  int r = row[e], c = col[e];
  float dx = pos[3*r+0] - pos[3*c+0];
  float dy = pos[3*r+1] - pos[3*c+1];
  float dz = pos[3*r+2] - pos[3*c+2];
  rbf[e] = __expf(-sqrtf(dx*dx + dy*dy + dz*dz));
}

// ---------------- agg = x  (residual init) ----------------
__global__ void copy_kernel(const float4* __restrict__ src, float4* __restrict__ dst) {
  int idx = blockIdx.x * blockDim.x + threadIdx.x;       // over N*64/4
  if (idx >= N_NODES * HIDDEN / 4) return;
  dst[idx] = src[idx];
}

// ---------------- agg[row[e]] += x[col[e]] * rbf[e] ----------------
// One wave32 per edge; edge index forced wave-uniform so row/col/rbf reads
// scalarize to s_load (SMEM). Each lane handles a float2 of the 64 features,
// scattered with native global_atomic_add_f32.
__global__ __launch_bounds__(256) void edge_kernel(
    const int* __restrict__ row, const int* __restrict__ col,
    const float* __restrict__ rbf, const float* __restrict__ x,
    float* __restrict__ agg) {
  // 256 threads = 8 waves per block; E = 1M = 125000 blocks * 8 exactly.
  int e = __builtin_amdgcn_readfirstlane((int)(blockIdx.x * 8 + (threadIdx.x >> 5)));
  int lane = threadIdx.x & 31;
  int r = row[e];
  int c = col[e];
  float s = rbf[e];
  int q = lane * 2;
  const float2 v = *(const float2*)(x + (size_t)c * HIDDEN + q);
  float* ad = agg + (size_t)r * HIDDEN + q;
  atomic_add_f32_global(ad + 0, v.x * s);
  atomic_add_f32_global(ad + 1, v.y * s);
}

// ---------------- Y = relu(X @ W^T + b), fp32 WMMA 16x16x4 ----------------
// 1 block = 16 nodes; 4 waves; wave w computes output cols [16w, 16w+16).
__global__ __launch_bounds__(128) void linear_wmma_kernel(
    const float* __restrict__ X, const float* __restrict__ W,
    const float* __restrict__ bias, float* __restrict__ Y) {
  int tile = blockIdx.x;               // node tile (N/16 = 6250)
  int wave = threadIdx.x >> 5;         // 0..3  -> N-tile of output
  int lane = threadIdx.x & 31;
  int lr   = lane & 15;
  int g    = lane >> 4;

  // A frag (16x4 f32): lane M = lr; VGPR j holds K = 2g + j  -> contiguous float2
  const float* Xrow = X + ((size_t)tile * 16 + lr) * HIDDEN;
  // B frag (4x16 f32): lane N = lr; VGPR j holds K = 2g + j; B[k][n] = W[n][k]
  const float* Wrow = W + (size_t)(wave * 16 + lr) * HIDDEN;

  v8f c = {};
#pragma unroll
  for (int kk = 0; kk < 16; ++kk) {
    int kb = kk * 4 + 2 * g;
    v2f a = *(const v2f*)(Xrow + kb);
    v2f b = *(const v2f*)(Wrow + kb);
    c = __builtin_amdgcn_wmma_f32_16x16x4_f32(
        /*neg_a=*/false, a, /*neg_b=*/false, b,
        /*c_mod=*/(short)0, c, /*reuse_a=*/false, /*reuse_b=*/false);
  }

  // D layout: lane n = lr; VGPR r -> row m = r + 8*g
  int ncol = wave * 16 + lr;
  float bi = bias[ncol];
  float* Yb = Y + (size_t)tile * 16 * HIDDEN + ncol;
#pragma unroll
  for (int r = 0; r < 8; ++r) {
    int m = r + 8 * g;
    float v = c[r] + bi;
    Yb[(size_t)m * HIDDEN] = v > 0.f ? v : 0.f;
  }
}

extern "C" void kernel_launch(void* const* d_in, const int* in_sizes, int n_in,
                              void* d_out, int out_size, void* d_ws, size_t ws_size,
                              hipStream_t stream) {
  const int*   z          = (const int*)d_in[0];
  const float* pos        = (const float*)d_in[1];
  const int*   edge_index = (const int*)d_in[2];
  const float* emb        = (const float*)d_in[3];
  const float* Ws         = (const float*)d_in[4];
  const float* bs         = (const float*)d_in[5];
  const int* row = edge_index;              // edge_index[0]
  const int* col = edge_index + N_EDGES;    // edge_index[1]

  float* x   = (float*)d_out;                          // N*64 (ping buffer; fully rewritten)
  float* agg = (float*)d_ws;                           // N*64
  float* rbf = agg + (size_t)N_NODES * HIDDEN;         // E

  embed_kernel<<<(N_NODES * HIDDEN + 255) / 256, 256, 0, stream>>>(z, emb, x);
  rbf_kernel<<<(N_EDGES + 255) / 256, 256, 0, stream>>>(pos, row, col, rbf);

  for (int l = 0; l < N_LAYERS; ++l) {
    copy_kernel<<<(N_NODES * HIDDEN / 4 + 255) / 256, 256, 0, stream>>>(
        (const float4*)x, (float4*)agg);
    edge_kernel<<<N_EDGES / 8, 256, 0, stream>>>(row, col, rbf, x, agg);
    linear_wmma_kernel<<<N_NODES / 16, 128, 0, stream>>>(
        agg, Ws + (size_t)l * HIDDEN * HIDDEN, bs + (size_t)l * HIDDEN, x);
  }
}